// Critic_3109556322709
// MI455X (gfx1250) — compile-verified
//
#include <hip/hip_runtime.h>
#include <hip/hip_bf16.h>
#include <math.h>

#define BATCH 256
#define NNODES 1000
#define HDIM 128

typedef float v2f __attribute__((ext_vector_type(2)));
typedef float v8f __attribute__((ext_vector_type(8)));

__device__ __forceinline__ float fast_tanh(float x) {
#if __has_builtin(__builtin_amdgcn_tanhf)
    return __builtin_amdgcn_tanhf(x);
#elif __has_builtin(__builtin_amdgcn_tanh_f32)
    return __builtin_amdgcn_tanh_f32(x);
#else
    float ax = fabsf(x);
    float e  = __expf(-2.0f * ax);
    float r  = (1.0f - e) / (1.0f + e);
    return copysignf(r, x);
#endif
}

// ---------------------------------------------------------------------------
// Fold rank-<=2 encoder structure into per-step 128-vectors:
// coeffs[kind][step][h], kind: 0=E0 1=E1 2=Ac 3=Ad 4=Ktanh 5=Khy
// ---------------------------------------------------------------------------
__global__ __launch_bounds__(HDIM) void coeff_kernel(
    const float* __restrict__ We_s, const float* __restrict__ be_s,
    const float* __restrict__ We_c, const float* __restrict__ be_c,
    const float* __restrict__ We_d, const float* __restrict__ be_d,
    const float* __restrict__ Wd,   const float* __restrict__ bd,
    const float* __restrict__ Wc,   const float* __restrict__ bc,
    const float* __restrict__ Wr,   const float* __restrict__ br,
    float* __restrict__ coeffs)
{
    const int i = blockIdx.x;    // step 0..2
    const int h = threadIdx.x;   // 0..127
    const float* wd = Wd + ((size_t)i * HDIM + h) * HDIM;
    const float* wc = Wc + ((size_t)i * HDIM + h) * HDIM;
    const float* wr = Wr + ((size_t)i * HDIM + h) * HDIM;
    float ad = 0.f, kd = 0.f, ac = 0.f, kc = 0.f, e0 = 0.f, e1 = 0.f, ke = 0.f;
    for (int k = 0; k < HDIM; ++k) {
        float wdk = wd[k], wck = wc[k], wrk = wr[k];
        ad = fmaf(wdk, We_d[k],       ad);
        kd = fmaf(wdk, be_d[k],       kd);
        ac = fmaf(wck, We_c[k],       ac);
        kc = fmaf(wck, be_c[k],       kc);
        e0 = fmaf(wrk, We_s[2*k + 0], e0);
        e1 = fmaf(wrk, We_s[2*k + 1], e1);
        ke = fmaf(wrk, be_s[k],       ke);
    }
    kd += bd[i * HDIM + h];
    kc += bc[i * HDIM + h];
    ke += br[i * HDIM + h];
    const int o = i * HDIM + h;
    coeffs[0 * 3 * HDIM + o] = e0;
    coeffs[1 * 3 * HDIM + o] = e1;
    coeffs[2 * 3 * HDIM + o] = ac;
    coeffs[3 * 3 * HDIM + o] = ad;
    coeffs[4 * 3 * HDIM + o] = kd + kc + ke;   // constant inside tanh
    coeffs[5 * 3 * HDIM + o] = kc + ke;        // constant of (e + c) for hy
}

__global__ void zero_kernel(float* __restrict__ p, int n) {
    int i = blockIdx.x * blockDim.x + threadIdx.x;
    if (i < n) p[i] = 0.f;
}

// ---------------------------------------------------------------------------
// Y[256,128] = X[256,128] @ W[128,128]^T + bias, optional ReLU.
// One wave per 16x16 output tile, V_WMMA_F32_16X16X4_F32, K-loop of 32.
// ---------------------------------------------------------------------------
__global__ __launch_bounds__(32) void gemm_wmma(
    const float* __restrict__ X, const float* __restrict__ W,
    const float* __restrict__ bias, float* __restrict__ Y, int do_relu)
{
    const int tile = blockIdx.x;
    const int m0 = (tile >> 3) * 16;       // 16 row tiles (batch)
    const int n0 = (tile & 7)  * 16;       // 8 col tiles (hidden)
    const int lane = threadIdx.x;          // 0..31, full wave, EXEC all ones
    const int r  = lane & 15;
    const int kb = (lane >> 4) * 2;

    v8f acc = {};
    const float* xrow = X + (size_t)(m0 + r) * HDIM;   // A: row m0+r
    const float* wrow = W + (size_t)(n0 + r) * HDIM;   // B: col n0+r of W^T = row of W
    for (int k0 = 0; k0 < HDIM; k0 += 4) {
        v2f a, b;
        a.x = xrow[k0 + kb];  a.y = xrow[k0 + kb + 1];
        b.x = wrow[k0 + kb];  b.y = wrow[k0 + kb + 1];
        acc = __builtin_amdgcn_wmma_f32_16x16x4_f32(
                  false, a, false, b, (short)0, acc, false, false);
    }

    const int col     = n0 + (lane & 15);
    const int rowbase = m0 + ((lane >> 4) ? 8 : 0);
    const float bv = bias[col];
#pragma unroll
    for (int rr = 0; rr < 8; ++rr) {
        float val = acc[rr] + bv;
        if (do_relu) val = fmaxf(val, 0.f);
        Y[(size_t)(rowbase + rr) * HDIM + col] = val;
    }
}

// ---------------------------------------------------------------------------
// One attention step: block = one batch element, 512 threads (16 wave32).
// Each lane owns h = lane + 32*j (j=0..3). LDS stages node arrays + logits.
// ---------------------------------------------------------------------------
__global__ __launch_bounds__(512) void attn_kernel(
    const float* __restrict__ statc,  // [B,2,N]
    const float* __restrict__ chl,    // [B,1,N]
    const float* __restrict__ dex,    // [B,1,N]
    const float* __restrict__ coeffs, // [6][3][128]
    const float* __restrict__ vvec,   // [3,128]
    const float* __restrict__ q,      // [B,128]
    float* __restrict__ hy,           // [B,128]
    int step)
{
    __shared__ float s0[NNODES], s1[NNODES], scl[NNODES], sdx[NNODES];
    __shared__ float su[NNODES];
    __shared__ float red[64];

    const int b    = blockIdx.x;
    const int tid  = threadIdx.x;
    const int lane = tid & 31;
    const int wv   = tid >> 5;
    const int nwv  = blockDim.x >> 5;   // 16 waves

    // Stage the four node arrays into LDS (16 KB).
    const float* pS0 = statc + (size_t)b * 2 * NNODES;
    const float* pS1 = pS0 + NNODES;
    const float* pCl = chl + (size_t)b * NNODES;
    const float* pDx = dex + (size_t)b * NNODES;
    for (int n = tid; n < NNODES; n += blockDim.x) {
        s0[n] = pS0[n]; s1[n] = pS1[n]; scl[n] = pCl[n]; sdx[n] = pDx[n];
    }

    // Per-lane coefficients for its 4 hidden slots.
    float e0[4], e1[4], ac[4], ad[4], kt[4], vv[4], qv[4];
#pragma unroll
    for (int j = 0; j < 4; ++j) {
        const int h = lane + 32 * j;
        const int o = step * HDIM + h;
        e0[j] = coeffs[0 * 3 * HDIM + o];
        e1[j] = coeffs[1 * 3 * HDIM + o];
        ac[j] = coeffs[2 * 3 * HDIM + o];
        ad[j] = coeffs[3 * 3 * HDIM + o];
        kt[j] = coeffs[4 * 3 * HDIM + o];
        vv[j] = vvec[o];
        qv[j] = q[(size_t)b * HDIM + h];
    }
    __syncthreads();

    // Logits: u[n] = sum_h v[h] * tanh(E0*s0 + E1*s1 + Ac*cl + Ad*dx + K + q[h])
    for (int n = wv; n < NNODES; n += nwv) {
        const float x0 = s0[n], x1 = s1[n], xc = scl[n], xd = sdx[n];
        float acc = 0.f;
#pragma unroll
        for (int j = 0; j < 4; ++j) {
            float a = fmaf(e0[j], x0,
                      fmaf(e1[j], x1,
                      fmaf(ac[j], xc,
                      fmaf(ad[j], xd, kt[j] + qv[j]))));
            acc = fmaf(vv[j], fast_tanh(a), acc);
        }
        for (int off = 16; off; off >>= 1) acc += __shfl_xor(acc, off, 32);
        if (lane == 0) su[n] = acc;
    }
    __syncthreads();

    // Block max over nodes.
    float m = -INFINITY;
    for (int n = tid; n < NNODES; n += blockDim.x) m = fmaxf(m, su[n]);
    for (int off = 16; off; off >>= 1) m = fmaxf(m, __shfl_xor(m, off, 32));
    if (lane == 0) red[wv] = m;
    __syncthreads();
    if (tid == 0) {
        float mm = red[0];
        for (int w = 1; w < nwv; ++w) mm = fmaxf(mm, red[w]);
        red[0] = mm;
    }
    __syncthreads();
    m = red[0];
    __syncthreads();

    // exp + the three weighted sums (unnormalized; divide by sumE at the end).
    float se = 0.f, a0 = 0.f, a1 = 0.f, acs = 0.f;
    for (int n = tid; n < NNODES; n += blockDim.x) {
        float e = __expf(su[n] - m);
        se += e;
        a0  = fmaf(e, s0[n],  a0);
        a1  = fmaf(e, s1[n],  a1);
        acs = fmaf(e, scl[n], acs);
    }
    for (int off = 16; off; off >>= 1) {
        se  += __shfl_xor(se,  off, 32);
        a0  += __shfl_xor(a0,  off, 32);
        a1  += __shfl_xor(a1,  off, 32);
        acs += __shfl_xor(acs, off, 32);
    }
    if (lane == 0) { red[wv] = se; red[16 + wv] = a0; red[32 + wv] = a1; red[48 + wv] = acs; }
    __syncthreads();
    if (tid == 0) {
        float t0 = 0.f, t1 = 0.f, t2 = 0.f, t3 = 0.f;
        for (int w = 0; w < nwv; ++w) {
            t0 += red[w]; t1 += red[16 + w]; t2 += red[32 + w]; t3 += red[48 + w];
        }
        red[0] = t0; red[16] = t1; red[32] = t2; red[48] = t3;
    }
    __syncthreads();
    const float inv = 1.0f / red[0];
    const float S0v = red[16] * inv, S1v = red[32] * inv, Scv = red[48] * inv;

    // hy[b,h] = E0*S0 + E1*S1 + Ac*Scl + (Kc + Ke)
    if (tid < HDIM) {
        const int o = step * HDIM + tid;
        const float E0v = coeffs[0 * 3 * HDIM + o];
        const float E1v = coeffs[1 * 3 * HDIM + o];
        const float Acv = coeffs[2 * 3 * HDIM + o];
        const float KHv = coeffs[5 * 3 * HDIM + o];
        hy[(size_t)b * HDIM + tid] =
            fmaf(E0v, S0v, fmaf(E1v, S1v, fmaf(Acv, Scv, KHv)));
    }
}

// out[b] = sum_h t[b,h] * W2[h] + b2   (t already has ReLU applied)
__global__ __launch_bounds__(HDIM) void head_kernel(
    const float* __restrict__ t, const float* __restrict__ W2,
    const float* __restrict__ b2, float* __restrict__ out)
{
    const int b = blockIdx.x;
    const int h = threadIdx.x;
    float v = t[(size_t)b * HDIM + h] * W2[h];
    for (int off = 16; off; off >>= 1) v += __shfl_xor(v, off, 32);
    __shared__ float red[4];
    if ((h & 31) == 0) red[h >> 5] = v;
    __syncthreads();
    if (h == 0) out[b] = red[0] + red[1] + red[2] + red[3] + b2[0];
}

extern "C" void kernel_launch(void* const* d_in, const int* in_sizes, int n_in,
                              void* d_out, int out_size, void* d_ws, size_t ws_size,
                              hipStream_t stream) {
    const float* statc = (const float*)d_in[0];
    const float* chl   = (const float*)d_in[1];
    const float* dex   = (const float*)d_in[2];
    const float* We_s  = (const float*)d_in[3];  const float* be_s = (const float*)d_in[4];
    const float* We_c  = (const float*)d_in[5];  const float* be_c = (const float*)d_in[6];
    const float* We_d  = (const float*)d_in[7];  const float* be_d = (const float*)d_in[8];
    const float* Wd    = (const float*)d_in[9];  const float* bd   = (const float*)d_in[10];
    const float* Wc    = (const float*)d_in[11]; const float* bc   = (const float*)d_in[12];
    const float* Wr    = (const float*)d_in[13]; const float* br   = (const float*)d_in[14];
    const float* Wq    = (const float*)d_in[15]; const float* bq   = (const float*)d_in[16];
    const float* vv    = (const float*)d_in[17];
    const float* W1    = (const float*)d_in[18]; const float* b1   = (const float*)d_in[19];
    const float* W2    = (const float*)d_in[20]; const float* b2   = (const float*)d_in[21];

    float* ws     = (float*)d_ws;
    float* coeffs = ws;                         // 6*3*128 = 2304 floats
    float* hy     = coeffs + 6 * 3 * HDIM;      // 256*128
    float* q      = hy + BATCH * HDIM;          // 256*128
    float* t      = q  + BATCH * HDIM;          // 256*128

    coeff_kernel<<<3, HDIM, 0, stream>>>(We_s, be_s, We_c, be_c, We_d, be_d,
                                         Wd, bd, Wc, bc, Wr, br, coeffs);
    zero_kernel<<<(BATCH * HDIM + 255) / 256, 256, 0, stream>>>(hy, BATCH * HDIM);

    for (int i = 0; i < 3; ++i) {
        gemm_wmma<<<(BATCH / 16) * (HDIM / 16), 32, 0, stream>>>(
            hy, Wq + (size_t)i * HDIM * HDIM, bq + i * HDIM, q, 0);
        attn_kernel<<<BATCH, 512, 0, stream>>>(statc, chl, dex, coeffs, vv, q, hy, i);
    }

    gemm_wmma<<<(BATCH / 16) * (HDIM / 16), 32, 0, stream>>>(hy, W1, b1, t, 1);
    head_kernel<<<BATCH, HDIM, 0, stream>>>(t, W2, b2, (float*)d_out);
}